// CPABActivationDifferent_53197464928907
// MI455X (gfx1250) — compile-verified
//
#include <hip/hip_runtime.h>
#include <hip/hip_bf16.h>

#define NCELL 16
#define KDIM  15     // NC-1
#define NCH   256
#define NPTS  131072
#define ROWS  32     // rows of x per block in the main kernel
#define TABF  (2 * NCELL * NCH)   // 8192 floats = 32KB: a-table then b-table

typedef __attribute__((ext_vector_type(2))) float v2f;
typedef __attribute__((ext_vector_type(8))) float v8f;

// ---------------------------------------------------------------------------
// Workspace layout (floats):
//   ws[0    .. 512)    : Bt padded to 16x32 (row k = basis vector k, row 15 = 0)
//   ws[512  .. 8704)   : transposed tables: a[cell][ch] (4096), b[cell][ch] (4096)
// ---------------------------------------------------------------------------

// Deterministic orthonormal null-space basis of the CPA constraint matrix:
// hat functions at internal knots, orthonormalized with Gram-Schmidt.
__global__ void gen_basis_kernel(float* __restrict__ ws_bt) {
    if (threadIdx.x != 0 || blockIdx.x != 0) return;
    float H[KDIM][2 * NCELL];
    for (int i = 0; i < KDIM; ++i)
        for (int j = 0; j < 2 * NCELL; ++j) H[i][j] = 0.0f;
    for (int k = 1; k <= KDIM; ++k) {
        float* h = H[k - 1];
        h[2 * (k - 1)]     =  16.0f;
        h[2 * (k - 1) + 1] = -(float)(k - 1);
        h[2 * k]           = -16.0f;
        h[2 * k + 1]       =  (float)(k + 1);
    }
    for (int i = 0; i < KDIM; ++i) {
        for (int j = 0; j < i; ++j) {
            float d = 0.0f;
            for (int t = 0; t < 2 * NCELL; ++t) d += H[i][t] * H[j][t];
            for (int t = 0; t < 2 * NCELL; ++t) H[i][t] -= d * H[j][t];
        }
        float n = 0.0f;
        for (int t = 0; t < 2 * NCELL; ++t) n += H[i][t] * H[i][t];
        float inv = rsqrtf(n);
        for (int t = 0; t < 2 * NCELL; ++t) H[i][t] *= inv;
    }
    for (int k = 0; k < 16; ++k)
        for (int c = 0; c < 32; ++c)
            ws_bt[k * 32 + c] = (k < KDIM) ? H[k][c] : 0.0f;
}

// A(256x32) = theta(256x15, K-padded to 16) @ Bt(16x32) via V_WMMA_F32_16X16X4_F32.
// 16 waves; wave w computes rows [16w,16w+16). Results are scattered directly
// into the TRANSPOSED table layout (a[cell][ch], b[cell][ch]) that the main
// kernel async-copies verbatim into LDS. Also copies theta to the output tail.
__global__ void __launch_bounds__(512)
proj_wmma_kernel(const float* __restrict__ theta,
                 const float* __restrict__ ws_bt,
                 float* __restrict__ ws_tab,
                 float* __restrict__ out_theta_tail) {
    const int t    = threadIdx.x;
    const int wave = t >> 5;        // 0..15
    const int lane = t & 31;
    const int half = lane >> 4;     // A-frag: lanes 0-15 hold K+0..1, 16-31 hold K+2..3
    const int l16  = lane & 15;

    // theta passthrough (second tuple element of the reference output)
    for (int i = t; i < NCH * KDIM; i += 512) out_theta_tail[i] = theta[i];

    const int mbase = wave * 16;
    v8f acc0 = {};
    v8f acc1 = {};
#pragma unroll
    for (int kk = 0; kk < 4; ++kk) {
        const int k0 = kk * 4 + half * 2;
        const int m  = mbase + l16;
        v2f afrag;
        afrag.x = (k0     < KDIM) ? theta[m * KDIM + k0]     : 0.0f;
        afrag.y = (k0 + 1 < KDIM) ? theta[m * KDIM + k0 + 1] : 0.0f;
        v2f b0, b1;
        b0.x = ws_bt[(k0)     * 32 + l16];
        b0.y = ws_bt[(k0 + 1) * 32 + l16];
        b1.x = ws_bt[(k0)     * 32 + 16 + l16];
        b1.y = ws_bt[(k0 + 1) * 32 + 16 + l16];
        acc0 = __builtin_amdgcn_wmma_f32_16x16x4_f32(false, afrag, false, b0,
                                                     (short)0, acc0, false, false);
        acc1 = __builtin_amdgcn_wmma_f32_16x16x4_f32(false, afrag, false, b1,
                                                     (short)0, acc1, false, false);
    }
    // C/D layout: VGPR r -> M(row/channel) = mbase + r + 8*half, N(col) = l16 / 16+l16.
    // col even -> a[col/2][ch]; col odd -> b[col/2][ch].
    float* aTab = ws_tab;                  // [16][256]
    float* bTab = ws_tab + NCELL * NCH;    // [16][256]
    const int col0 = l16;
    const int col1 = 16 + l16;
    float* base0 = (col0 & 1) ? bTab : aTab;
    float* base1 = (col1 & 1) ? bTab : aTab;
#pragma unroll
    for (int r = 0; r < 8; ++r) {
        const int ch = mbase + r + half * 8;
        base0[(col0 >> 1) * NCH + ch] = acc0[r];
        base1[(col1 >> 1) * NCH + ch] = acc1[r];
    }
}

// Main element-wise CPAB integration. The 32KB (a,b) table is streamed into
// LDS with async b128 global->LDS transfers (ASYNCcnt), then gathered
// bank-conflict-free (lane = channel, 256 % 64banks == 0).
__global__ void __launch_bounds__(256)
cpab_main_kernel(const float* __restrict__ x,
                 const float* __restrict__ ws_tab,
                 const int* __restrict__ time_p,
                 float* __restrict__ out) {
    __shared__ float sTab[TABF];           // a-table [16][256] then b-table [16][256]
    const int c = threadIdx.x;             // channel 0..255

    // Async copy: 8192 floats = 2048 x b128; 256 threads x 8 iterations.
#pragma unroll
    for (int i = 0; i < 8; ++i) {
        const unsigned f = (unsigned)(i * 1024 + c * 4);            // float index
        const unsigned gbyte = f * 4u;                              // global byte offset
        const unsigned lbyte = (unsigned)(size_t)(&sTab[f]);        // LDS byte address
        asm volatile("global_load_async_to_lds_b128 %0, %1, %2"
                     :
                     : "v"(lbyte), "v"(gbyte), "s"(ws_tab)
                     : "memory");
    }
    asm volatile("s_wait_asynccnt 0x0" ::: "memory");
    __syncthreads();

    const float* sA = sTab;
    const float* sB = sTab + NCELL * NCH;

    const float tme = (float)(*time_p);
    const float dt  = tme * 0.1f;          // time / NSTEPS1
    const float ddt = dt * 0.2f;           // dt / NSTEPS2

    const long long base = (long long)blockIdx.x * ROWS * NCH + c;

    for (int r = 0; r < ROWS; ++r) {
        const long long idx = base + (long long)r * NCH;
        if (r + 4 < ROWS) __builtin_prefetch(&x[idx + 4 * NCH], 0, 0);
        const float xv = x[idx];
        const float xs = (xv + 3.0f) * (1.0f / 6.0f);
        const bool ood = (xs >= 1.0f) | (xs <= 0.0f);

        float p = xs;
#pragma unroll 1
        for (int s = 0; s < 10; ++s) {
            int c0 = (int)floorf(p * 16.0f);
            c0 = c0 < 0 ? 0 : (c0 > 15 ? 15 : c0);
            const float ac = sA[c0 * NCH + c];
            const float bc = sB[c0 * NCH + c];
            const float eta = __expf(dt * ac);
            const float pcf = (fabsf(ac) > 1e-7f)
                                  ? (eta * p + (bc / ac) * (eta - 1.0f))
                                  : (p + bc * dt);
            int c1 = (int)floorf(pcf * 16.0f);
            c1 = c1 < 0 ? 0 : (c1 > 15 ? 15 : c1);
            if (c1 == c0) {
                p = pcf;
            } else {
                float q = p;
#pragma unroll
                for (int ss = 0; ss < 5; ++ss) {
                    int cq = (int)floorf(q * 16.0f);
                    cq = cq < 0 ? 0 : (cq > 15 ? 15 : cq);
                    q += ddt * (sA[cq * NCH + c] * q + sB[cq * NCH + c]);
                }
                p = q;
            }
        }
        const float res = p * 6.0f - 3.0f;
        out[idx] = ood ? xv : res;
    }
}

extern "C" void kernel_launch(void* const* d_in, const int* in_sizes, int n_in,
                              void* d_out, int out_size, void* d_ws, size_t ws_size,
                              hipStream_t stream) {
    const float* x      = (const float*)d_in[0];   // (131072, 256)
    const int*   time_p = (const int*)d_in[4];     // scalar 1
    const float* theta  = (const float*)d_in[5];   // (256, 15)
    float* out = (float*)d_out;                    // 131072*256 + 256*15 floats
    float* ws  = (float*)d_ws;

    float* ws_bt  = ws;        // 16 x 32
    float* ws_tab = ws + 512;  // a[16][256] + b[16][256]

    gen_basis_kernel<<<1, 32, 0, stream>>>(ws_bt);
    proj_wmma_kernel<<<1, 512, 0, stream>>>(theta, ws_bt, ws_tab,
                                            out + (size_t)NPTS * NCH);
    cpab_main_kernel<<<NPTS / ROWS, 256, 0, stream>>>(x, ws_tab, time_p, out);
}